// SingleImageRasterization_85091892068504
// MI455X (gfx1250) — compile-verified
//
#include <hip/hip_runtime.h>
#include <math.h>

// ---------------------------------------------------------------------------
// Single-image 3D Gaussian splat rasterizer for MI455X (gfx1250, wave32).
// Compositing accumulation runs on V_WMMA_F32_16X16X4_F32: each wave owns 16
// pixels (C rows) and 16 output channels (C cols; 0..4 = r,g,b,acc,depth).
// ---------------------------------------------------------------------------

#define HH 128
#define WW 128
#define NPAD 2048
#define CHUNK 256
#define ZNEARF 0.01f
#define ZFARF 100.0f
#define EPS2DF 0.3f
#define ALPHA_CLIPF 0.999f
#define ALPHA_MINF (1.0f / 255.0f)

typedef __attribute__((ext_vector_type(2))) float v2f;
typedef __attribute__((ext_vector_type(8))) float v8f;

// ---------------------------------------------------------------------------
// Kernel 1: per-gaussian projection / 2D covariance. One thread per gaussian.
// Record layout (12 floats): mx,my,cA,cB,cC,opac,r,g,b,z,0,0
// ---------------------------------------------------------------------------
__global__ void prep_kernel(const float* __restrict__ means,
                            const float* __restrict__ quats,
                            const float* __restrict__ scales,
                            const float* __restrict__ opacs,
                            const float* __restrict__ colors,
                            const float* __restrict__ vm,
                            const float* __restrict__ Km,
                            float* __restrict__ rec,
                            float* __restrict__ keys, int N) {
  int i = blockIdx.x * blockDim.x + threadIdx.x;
  if (i >= NPAD) return;
  if (i >= N) {
    keys[i] = __builtin_inff();
#pragma unroll
    for (int q = 0; q < 12; q++) rec[i * 12 + q] = 0.0f;
    return;
  }
  float m0 = means[3 * i], m1 = means[3 * i + 1], m2 = means[3 * i + 2];
  float R00 = vm[0], R01 = vm[1], R02 = vm[2], T0 = vm[3];
  float R10 = vm[4], R11 = vm[5], R12 = vm[6], T1 = vm[7];
  float R20 = vm[8], R21 = vm[9], R22 = vm[10], T2 = vm[11];
  float x = R00 * m0 + R01 * m1 + R02 * m2 + T0;
  float y = R10 * m0 + R11 * m1 + R12 * m2 + T1;
  float z = R20 * m0 + R21 * m1 + R22 * m2 + T2;
  bool valid = (z > ZNEARF) && (z < ZFARF);
  float zc = fmaxf(z, 1e-6f);
  float fx = Km[0], cx = Km[2], fy = Km[4], cy = Km[5];

  // quat -> rotmat, scaled columns
  float qw = quats[4 * i], qx = quats[4 * i + 1], qy = quats[4 * i + 2],
        qz = quats[4 * i + 3];
  float qn = rsqrtf(qw * qw + qx * qx + qy * qy + qz * qz);
  qw *= qn; qx *= qn; qy *= qn; qz *= qn;
  float s0 = scales[3 * i], s1 = scales[3 * i + 1], s2 = scales[3 * i + 2];
  float Q00 = 1.0f - 2.0f * (qy * qy + qz * qz), Q01 = 2.0f * (qx * qy - qw * qz),
        Q02 = 2.0f * (qx * qz + qw * qy);
  float Q10 = 2.0f * (qx * qy + qw * qz), Q11 = 1.0f - 2.0f * (qx * qx + qz * qz),
        Q12 = 2.0f * (qy * qz - qw * qx);
  float Q20 = 2.0f * (qx * qz - qw * qy), Q21 = 2.0f * (qy * qz + qw * qx),
        Q22 = 1.0f - 2.0f * (qx * qx + qy * qy);
  float M00 = Q00 * s0, M01 = Q01 * s1, M02 = Q02 * s2;
  float M10 = Q10 * s0, M11 = Q11 * s1, M12 = Q12 * s2;
  float M20 = Q20 * s0, M21 = Q21 * s1, M22 = Q22 * s2;

  // cov3 = M M^T (symmetric)
  float C00 = M00 * M00 + M01 * M01 + M02 * M02;
  float C01 = M00 * M10 + M01 * M11 + M02 * M12;
  float C02 = M00 * M20 + M01 * M21 + M02 * M22;
  float C11 = M10 * M10 + M11 * M11 + M12 * M12;
  float C12 = M10 * M20 + M11 * M21 + M12 * M22;
  float C22 = M20 * M20 + M21 * M21 + M22 * M22;

  // covc = R cov3 R^T ; W = R*cov3
  float W00 = R00 * C00 + R01 * C01 + R02 * C02;
  float W01 = R00 * C01 + R01 * C11 + R02 * C12;
  float W02 = R00 * C02 + R01 * C12 + R02 * C22;
  float W10 = R10 * C00 + R11 * C01 + R12 * C02;
  float W11 = R10 * C01 + R11 * C11 + R12 * C12;
  float W12 = R10 * C02 + R11 * C12 + R12 * C22;
  float W20 = R20 * C00 + R21 * C01 + R22 * C02;
  float W21 = R20 * C01 + R21 * C11 + R22 * C12;
  float W22 = R20 * C02 + R21 * C12 + R22 * C22;
  float V00 = W00 * R00 + W01 * R01 + W02 * R02;
  float V01 = W00 * R10 + W01 * R11 + W02 * R12;
  float V02 = W00 * R20 + W01 * R21 + W02 * R22;
  float V11 = W10 * R10 + W11 * R11 + W12 * R12;
  float V12 = W10 * R20 + W11 * R21 + W12 * R22;
  float V22 = W20 * R20 + W21 * R21 + W22 * R22;

  float iz = 1.0f / zc;
  float j00 = fx * iz, j02 = -fx * x * iz * iz;
  float j11 = fy * iz, j12 = -fy * y * iz * iz;
  // U = J * covc (2x3)
  float U00 = j00 * V00 + j02 * V02;
  float U01 = j00 * V01 + j02 * V12;
  float U02 = j00 * V02 + j02 * V22;
  float U10 = j11 * V01 + j12 * V02;
  float U11 = j11 * V11 + j12 * V12;
  float U12 = j11 * V12 + j12 * V22;
  float a = U00 * j00 + U02 * j02 + EPS2DF;
  float b = U01 * j11 + U02 * j12;
  float c = U11 * j11 + U12 * j12 + EPS2DF;
  float idet = 1.0f / (a * c - b * b);

  float* r = rec + i * 12;
  r[0] = fx * x * iz + cx;       // mx
  r[1] = fy * y * iz + cy;       // my
  r[2] = c * idet;               // cA
  r[3] = -b * idet;              // cB
  r[4] = a * idet;               // cC
  r[5] = valid ? opacs[i] : 0.0f;
  r[6] = colors[3 * i];
  r[7] = colors[3 * i + 1];
  r[8] = colors[3 * i + 2];
  r[9] = z;
  r[10] = 0.0f;
  r[11] = 0.0f;
  keys[i] = valid ? z : __builtin_inff();
}

// ---------------------------------------------------------------------------
// Kernel 2: single-block LDS bitonic sort of 2048 (z,idx) pairs + gather into
//   geo  : 8 floats/gaussian  (mx,my,cA,cB,cC,opac,0,0)
//   attr : 16 floats/gaussian (r,g,b,1,z,0,...,0)  == one B-matrix row
// ---------------------------------------------------------------------------
__global__ void __launch_bounds__(1024) sort_gather_kernel(
    const float* __restrict__ keys, const float* __restrict__ rec,
    float* __restrict__ geo, float* __restrict__ attr) {
  __shared__ float sk[NPAD];
  __shared__ int sv[NPAD];
  int t = threadIdx.x;
  for (int i = t; i < NPAD; i += 1024) {
    sk[i] = keys[i];
    sv[i] = i;
  }
  __syncthreads();
  for (int k = 2; k <= NPAD; k <<= 1) {
    for (int j = k >> 1; j > 0; j >>= 1) {
      for (int s = t; s < NPAD; s += 1024) {
        int ixj = s ^ j;
        if (ixj > s) {
          bool up = ((s & k) == 0);
          float ka = sk[s], kb = sk[ixj];
          bool doswap = up ? (ka > kb) : (ka < kb);
          if (doswap) {
            sk[s] = kb; sk[ixj] = ka;
            int tv = sv[s]; sv[s] = sv[ixj]; sv[ixj] = tv;
          }
        }
      }
      __syncthreads();
    }
  }
  for (int s = t; s < NPAD; s += 1024) {
    const float* r = rec + (size_t)sv[s] * 12;
    float* g = geo + (size_t)s * 8;
    g[0] = r[0]; g[1] = r[1]; g[2] = r[2]; g[3] = r[3];
    g[4] = r[4]; g[5] = r[5]; g[6] = 0.0f; g[7] = 0.0f;
    float* at = attr + (size_t)s * 16;
    at[0] = r[6]; at[1] = r[7]; at[2] = r[8];
    at[3] = 1.0f; at[4] = r[9];
#pragma unroll
    for (int q = 5; q < 16; q++) at[q] = 0.0f;
  }
}

// ---------------------------------------------------------------------------
// Kernel 3: rasterize. 8 waves/block, one wave = 16 consecutive pixels.
// Inner loop: 4 gaussians/step -> V_WMMA_F32_16X16X4_F32 accumulate of
// w[16px x 4g] x attr[4g x 16chan].
// ---------------------------------------------------------------------------
__global__ void __launch_bounds__(256) raster_kernel(
    const float* __restrict__ geo, const float* __restrict__ attr,
    float* __restrict__ out) {
  __shared__ __align__(16) float sgeo[CHUNK * 8];    // 8 KB
  __shared__ __align__(16) float sattr[CHUNK * 16];  // 16 KB
  int tid = threadIdx.x;
  int lane = tid & 31;
  int wave = tid >> 5;
  int gw = blockIdx.x * 8 + wave;
  int pixbase = gw * 16;             // 16 consecutive pixels in one row
  int mrow = lane & 15;              // C-matrix column (channel) / A row (pixel)
  int h = lane >> 4;                 // wave half: owns K pair {0,1} or {2,3}
  float px = (float)((pixbase & (WW - 1)) + mrow) + 0.5f;
  float py = (float)(pixbase >> 7) + 0.5f;

  v8f acc = {0.f, 0.f, 0.f, 0.f, 0.f, 0.f, 0.f, 0.f};
  float T = 1.0f;

  for (int base = 0; base < NPAD; base += CHUNK) {
    __syncthreads();
    {
      const float4* gs = (const float4*)(geo + (size_t)(base + tid) * 8);
      float4* gd = (float4*)(sgeo + tid * 8);
      gd[0] = gs[0]; gd[1] = gs[1];
      const float4* as = (const float4*)(attr + (size_t)(base + tid) * 16);
      float4* ad = (float4*)(sattr + tid * 16);
      ad[0] = as[0]; ad[1] = as[1]; ad[2] = as[2]; ad[3] = as[3];
      if (base + CHUNK < NPAD) {  // head start on next chunk (global_prefetch)
        __builtin_prefetch(geo + (size_t)(base + CHUNK + tid) * 8, 0, 1);
        __builtin_prefetch(attr + (size_t)(base + CHUNK + tid) * 16, 0, 1);
      }
    }
    __syncthreads();

    for (int sub = 0; sub < CHUNK; sub += 4) {
      int gA = sub + 2 * h;                 // this half's gaussian pair
      const float* pA = sgeo + gA * 8;
      const float* pB = pA + 8;

      float dxA = px - pA[0], dyA = py - pA[1];
      float sA = 0.5f * (pA[2] * dxA * dxA + pA[4] * dyA * dyA) + pA[3] * dxA * dyA;
      float alA = fminf(ALPHA_CLIPF, pA[5] * __expf(-sA));
      alA = (sA >= 0.0f && alA > ALPHA_MINF) ? alA : 0.0f;

      float dxB = px - pB[0], dyB = py - pB[1];
      float sB = 0.5f * (pB[2] * dxB * dxB + pB[4] * dyB * dyB) + pB[3] * dxB * dyB;
      float alB = fminf(ALPHA_CLIPF, pB[5] * __expf(-sB));
      alB = (sB >= 0.0f && alB > ALPHA_MINF) ? alB : 0.0f;

      // pair transmittance product; one exchange between wave halves
      float qown = (1.0f - alA) * (1.0f - alB);
      float qoth = __shfl_xor(qown, 16, 32);

      // lanes owning K pair {2,3} (h==1) are preceded by the other pair
      float Tpre = T * (h ? qoth : 1.0f);
      float w0 = Tpre * alA;
      float t1 = Tpre - w0;
      float w1 = t1 * alB;
      T = T * qown * qoth;  // identical update in both halves

      // A matrix: lanes 0-15 hold K=0,1 ; lanes 16-31 hold K=2,3 (ISA layout)
      v2f av; av.x = w0; av.y = w1;

      // B matrix row fetch: 64 consecutive floats across the wave, no
      // bank conflicts (attr[g][chan], chan = lane&15)
      v2f bv;
      bv.x = sattr[gA * 16 + mrow];
      bv.y = sattr[gA * 16 + 16 + mrow];

      acc = __builtin_amdgcn_wmma_f32_16x16x4_f32(
          /*neg_a=*/false, av, /*neg_b=*/false, bv,
          /*c_mod=*/(short)0, acc, /*reuse_a=*/false, /*reuse_b=*/false);
    }
  }

  // C layout: VGPR j holds row M = j + 8*h, lane column N = mrow.
  float* img = out;                 // (H,W,3)
  float* acca = out + HH * WW * 3;  // (H,W,1)
  float* dpt = out + HH * WW * 4;   // (H,W,1)
#pragma unroll
  for (int j = 0; j < 8; j++) {
    float v = acc[j];
    float accv = __shfl(v, (lane & 16) | 3, 32);  // channel-3 (acc_a) value
    int pix = pixbase + j + (h ? 8 : 0);
    if (mrow < 3) {
      img[pix * 3 + mrow] = v;
    } else if (mrow == 3) {
      acca[pix] = v;
    } else if (mrow == 4) {
      dpt[pix] = v / fmaxf(accv, 1e-10f);
    }
  }
}

// ---------------------------------------------------------------------------
extern "C" void kernel_launch(void* const* d_in, const int* in_sizes, int n_in,
                              void* d_out, int out_size, void* d_ws,
                              size_t ws_size, hipStream_t stream) {
  const float* means = (const float*)d_in[0];
  const float* quats = (const float*)d_in[1];
  const float* scales = (const float*)d_in[2];
  const float* opacs = (const float*)d_in[3];
  const float* colors = (const float*)d_in[4];
  const float* vm = (const float*)d_in[5];
  const float* Km = (const float*)d_in[6];
  int N = in_sizes[3];  // opacities: flat count == N gaussians

  float* rec = (float*)d_ws;            // NPAD*12 floats
  float* keys = rec + NPAD * 12;        // NPAD floats
  float* geo = keys + NPAD;             // NPAD*8 floats
  float* attr = geo + NPAD * 8;         // NPAD*16 floats

  prep_kernel<<<NPAD / 256, 256, 0, stream>>>(means, quats, scales, opacs,
                                              colors, vm, Km, rec, keys, N);
  sort_gather_kernel<<<1, 1024, 0, stream>>>(keys, rec, geo, attr);
  raster_kernel<<<(HH * WW) / (8 * 16), 256, 0, stream>>>(geo, attr,
                                                          (float*)d_out);
}